// GatedMessageGcn_3126736191774
// MI455X (gfx1250) — compile-verified
//
#include <hip/hip_runtime.h>

typedef __attribute__((ext_vector_type(2))) float v2f;
typedef __attribute__((ext_vector_type(8))) float v8f;

#define R_DIM 1000
#define D_DIM 100
#define M_SEG 2000000

// ---------------------------------------------------------------------------
// Kernel 1: energies = G_sender (R x D) * G_receiver^T (D x R), fp32 WMMA.
// One wave (32 threads) computes one 16x16 tile via V_WMMA_F32_16X16X4_F32,
// accumulating over D=100 in K-steps of 4 (25 WMMA ops per wave).
//
// A 16x4 f32 layout : lanes 0-15 -> M=lane, K={0,1}; lanes 16-31 -> M=lane-16, K={2,3}
// B 4x16 f32 layout : lanes 0-15 -> N=lane, K={0,1}; lanes 16-31 -> N=lane-16, K={2,3}
// C 16x16 f32 layout: VGPR v, lanes 0-15 -> M=v, N=lane; lanes 16-31 -> M=v+8, N=lane-16
// Loads are clamped (never OOB, EXEC stays all-1s through the WMMA loop);
// stores are bounds-guarded.
// ---------------------------------------------------------------------------
__global__ __launch_bounds__(32) void gemm_energies_wmma(
    const float* __restrict__ GS, const float* __restrict__ GR,
    float* __restrict__ Emat) {
  const int tileM = blockIdx.y * 16;
  const int tileN = blockIdx.x * 16;
  const int lane  = threadIdx.x;
  const int half  = lane >> 4;    // 0 or 1
  const int l15   = lane & 15;

  int m = tileM + l15; m = (m < R_DIM) ? m : (R_DIM - 1);
  int n = tileN + l15; n = (n < R_DIM) ? n : (R_DIM - 1);
  const int koff = half * 2;

  const float* __restrict__ arow = GS + (size_t)m * D_DIM + koff;
  const float* __restrict__ brow = GR + (size_t)n * D_DIM + koff;

  v8f c = {};
#pragma unroll 5
  for (int k = 0; k < D_DIM; k += 4) {
    v2f a, b;
    a.x = arow[k];
    a.y = arow[k + 1];
    b.x = brow[k];
    b.y = brow[k + 1];
    // 8 args: (neg_a, A, neg_b, B, c_mod, C, reuse_a, reuse_b)
    c = __builtin_amdgcn_wmma_f32_16x16x4_f32(
        false, a, false, b, (short)0, c, false, false);
  }

#pragma unroll
  for (int v = 0; v < 8; ++v) {
    const int row = tileM + v + half * 8;
    const int col = tileN + l15;
    if (row < R_DIM && col < R_DIM)
      Emat[(size_t)row * R_DIM + col] = c[v];
  }
}

// ---------------------------------------------------------------------------
// Kernel 2: zero the segment-sum accumulator (M floats, ~8MB, L2-resident).
// M is read from the device scalar so we never dereference device memory on
// the host (graph-capture safe).
// ---------------------------------------------------------------------------
__global__ void zero_denom(float* __restrict__ denom,
                           const int* __restrict__ nm) {
  const int total  = *nm;
  const int stride = gridDim.x * blockDim.x;
  for (int i = blockIdx.x * blockDim.x + threadIdx.x; i < total; i += stride)
    denom[i] = 0.0f;
}

// ---------------------------------------------------------------------------
// Kernel 3: per edge -> gather energy, sigmoid, stash gate, atomically
// accumulate denom[receiver]. denom is 8MB -> atomics resolve in L2.
// ---------------------------------------------------------------------------
__global__ __launch_bounds__(256) void edge_gates(
    const int* __restrict__ rel, const int* __restrict__ pairs,
    const float* __restrict__ Emat, float* __restrict__ gates,
    float* __restrict__ denom, int E) {
  const int e = blockIdx.x * blockDim.x + threadIdx.x;
  if (e >= E) return;
  const int2 rp = ((const int2*)rel)[e];               // (sender_rel, receiver_rel)
  const float x = Emat[(size_t)rp.x * R_DIM + rp.y];   // L2-resident gather (4MB)
  const float g = 1.0f / (1.0f + __expf(-x));          // v_exp_f32 path
  gates[e] = g;
  const int recv = ((const int2*)pairs)[e].y;
  atomicAdd(denom + recv, g);
}

// ---------------------------------------------------------------------------
// Kernel 4: per edge -> weight = gate / (denom[receiver] + 1e-8).
// ---------------------------------------------------------------------------
__global__ __launch_bounds__(256) void edge_weights(
    const int* __restrict__ pairs, const float* __restrict__ gates,
    const float* __restrict__ denom, float* __restrict__ out, int E) {
  const int e = blockIdx.x * blockDim.x + threadIdx.x;
  if (e >= E) return;
  const int recv = ((const int2*)pairs)[e].y;
  out[e] = gates[e] / (denom[recv] + 1e-8f);
}

// ---------------------------------------------------------------------------
// Launcher. Workspace layout (76MB total):
//   [0, 4MB)        Emat   : 1000*1000 f32
//   [4MB, 12MB)     denom  : 2,000,000 f32
//   [12MB, 76MB)    gates  : 16,000,000 f32
// ---------------------------------------------------------------------------
extern "C" void kernel_launch(void* const* d_in, const int* in_sizes, int n_in,
                              void* d_out, int out_size, void* d_ws,
                              size_t ws_size, hipStream_t stream) {
  const float* GS    = (const float*)d_in[0];
  const float* GR    = (const float*)d_in[1];
  const int*   rel   = (const int*)d_in[2];
  const int*   pairs = (const int*)d_in[3];
  const int*   nm    = (const int*)d_in[4];
  float*       out   = (float*)d_out;

  const int E = in_sizes[2] / 2;

  char*  ws    = (char*)d_ws;
  float* Emat  = (float*)(ws);
  float* denom = (float*)(ws + (size_t)R_DIM * R_DIM * sizeof(float));
  float* gates = (float*)(ws + (size_t)R_DIM * R_DIM * sizeof(float) +
                          (size_t)M_SEG * sizeof(float));

  const int ntiles = (R_DIM + 15) / 16;  // 63
  dim3 ggrid(ntiles, ntiles);
  gemm_energies_wmma<<<ggrid, 32, 0, stream>>>(GS, GR, Emat);

  zero_denom<<<2048, 256, 0, stream>>>(denom, nm);

  const int eblocks = (E + 255) / 256;
  edge_gates<<<eblocks, 256, 0, stream>>>(rel, pairs, Emat, gates, denom, E);
  edge_weights<<<eblocks, 256, 0, stream>>>(pairs, gates, denom, out, E);
}